// StructuredTernaryFactorGenerator_73280732004900
// MI455X (gfx1250) — compile-verified
//
#include <hip/hip_runtime.h>
#include <hip/hip_bf16.h>
#include <math.h>

// ---------------------------------------------------------------------------
// StructuredTernaryFactorGenerator for MI455X (gfx1250, wave32, WMMA)
//   B=64, H=1024, O=1024, I=1024, K=8
// Phase 1: three MLPs via V_WMMA_F32_16X16X4_F32 (full f32 precision on the
//          matrix pipe; 256 wmma ops per 16x16 output tile).
// Phase 2: bandwidth-bound weight construction; bases/base tiles cached in
//          registers across the batch loop so global reads stay at ~36 MB and
//          the 256 MB output streams out with non-temporal b128 stores.
// ---------------------------------------------------------------------------

typedef __attribute__((ext_vector_type(2))) float v2f;
typedef __attribute__((ext_vector_type(4))) float v4f;
typedef __attribute__((ext_vector_type(8))) float v8f;

#define HDIM 1024
#define ODIM 1024
#define IDIM 1024
#define BDIM 64
#define KDIM 8

// ---------------------------------------------------------------------------
// Wave-level 16x16 f32 GEMM tile: C[16,16] = A[m0:m0+16, :K] * W[:K, n0:n0+16]
// A row-major (lda), W row-major (ldb).  wave32 WMMA layout:
//   A frag (v2f): lane L holds A[m0 + L%16][k + 2*(L/16) + v]
//   B frag (v2f): lane L holds W[k + 2*(L/16) + v][n0 + L%16]
//   C frag (v8f): lane L, vgpr v holds C[m0 + 8*(L/16) + v][n0 + L%16]
// ---------------------------------------------------------------------------
__device__ __forceinline__ v8f wave_gemm_tile_f32(const float* __restrict__ A,
                                                  int lda,
                                                  const float* __restrict__ W,
                                                  int ldb,
                                                  int m0, int n0, int K,
                                                  int lane) {
  const int r    = lane & 15;
  const int half = lane >> 4;
  v8f acc = {0.f, 0.f, 0.f, 0.f, 0.f, 0.f, 0.f, 0.f};
  const float* __restrict__ Arow = A + (size_t)(m0 + r) * lda;
  const float* __restrict__ Bcol = W + (n0 + r);
  for (int k = 0; k < K; k += 4) {
    const int ka = k + 2 * half;
    const float2 av = *(const float2*)(Arow + ka);      // 8B-aligned b64 load
    v2f a; a[0] = av.x; a[1] = av.y;
    v2f b;
    b[0] = Bcol[(size_t)ka * ldb];
    b[1] = Bcol[(size_t)(ka + 1) * ldb];
    acc = __builtin_amdgcn_wmma_f32_16x16x4_f32(
        /*neg_a=*/false, a, /*neg_b=*/false, b,
        /*c_mod=*/(short)0, acc, /*reuse_a=*/false, /*reuse_b=*/false);
  }
  return acc;
}

// ---------------------------------------------------------------------------
// Layer 1 (x3 via gridDim.z): t = silu(h @ W1 + b1),  [64,1024]x[1024,1024]
// 4 waves / block, 1 tile / wave, 64 blocks -> 256 tiles (4 x 64 of 16x16).
// ---------------------------------------------------------------------------
__global__ __launch_bounds__(128) void mlp_layer1_kernel(
    const float* __restrict__ h,
    const float* __restrict__ rw1, const float* __restrict__ rb1,
    const float* __restrict__ cw1, const float* __restrict__ cb1,
    const float* __restrict__ kw1, const float* __restrict__ kb1,
    float* __restrict__ t1, float* __restrict__ t2, float* __restrict__ t3) {
  const float* W; const float* bias; float* out;
  if (blockIdx.z == 0)      { W = rw1; bias = rb1; out = t1; }
  else if (blockIdx.z == 1) { W = cw1; bias = cb1; out = t2; }
  else                      { W = kw1; bias = kb1; out = t3; }

  const int wave = threadIdx.x >> 5;
  const int lane = threadIdx.x & 31;
  const int tile = blockIdx.x * 4 + wave;   // [0,256)
  const int m0 = (tile >> 6) << 4;          // 4 M-tiles
  const int n0 = (tile & 63) << 4;          // 64 N-tiles

  v8f acc = wave_gemm_tile_f32(h, HDIM, W, HDIM, m0, n0, HDIM, lane);

  const int r    = lane & 15;
  const int half = lane >> 4;
  const float bv = bias[n0 + r];
#pragma unroll
  for (int v = 0; v < 8; ++v) {
    const int m = m0 + 8 * half + v;
    const float x = acc[v] + bv;
    const float s = x / (1.f + expf(-x));   // SiLU
    out[(size_t)m * HDIM + n0 + r] = s;
  }
}

// ---------------------------------------------------------------------------
// Layer 2 for row/col (x2 via gridDim.z): y = 1 + 0.1*tanh(t @ W2 + b2)
// ---------------------------------------------------------------------------
__global__ __launch_bounds__(128) void mlp_layer2_kernel(
    const float* __restrict__ t1, const float* __restrict__ t2,
    const float* __restrict__ rw2, const float* __restrict__ rb2,
    const float* __restrict__ cw2, const float* __restrict__ cb2,
    float* __restrict__ row_out, float* __restrict__ col_out) {
  const float* T; const float* W; const float* bias; float* out;
  if (blockIdx.z == 0) { T = t1; W = rw2; bias = rb2; out = row_out; }
  else                 { T = t2; W = cw2; bias = cb2; out = col_out; }

  const int wave = threadIdx.x >> 5;
  const int lane = threadIdx.x & 31;
  const int tile = blockIdx.x * 4 + wave;
  const int m0 = (tile >> 6) << 4;
  const int n0 = (tile & 63) << 4;

  v8f acc = wave_gemm_tile_f32(T, HDIM, W, HDIM, m0, n0, HDIM, lane);

  const int r    = lane & 15;
  const int half = lane >> 4;
  const float bv = bias[n0 + r];
#pragma unroll
  for (int v = 0; v < 8; ++v) {
    const int m = m0 + 8 * half + v;
    const float x = acc[v] + bv;
    out[(size_t)m * HDIM + n0 + r] = 1.f + 0.1f * tanhf(x);
  }
}

// ---------------------------------------------------------------------------
// coeff = tanh(t3 @ kw2 + kb2),  [64,1024]x[1024,8].  N=8 padded to a 16-wide
// WMMA tile; B-fragment loads guarded by ternaries (no EXEC divergence around
// the WMMA), stores masked to n<8.  One block, 4 waves (one per 16-row M tile).
// ---------------------------------------------------------------------------
__global__ __launch_bounds__(128) void mlp_coeff_kernel(
    const float* __restrict__ t3,
    const float* __restrict__ kw2, const float* __restrict__ kb2,
    float* __restrict__ coeff_out) {
  const int wave = threadIdx.x >> 5;
  const int lane = threadIdx.x & 31;
  const int r    = lane & 15;
  const int half = lane >> 4;
  const int m0   = wave << 4;
  const bool nvalid = (r < KDIM);

  v8f acc = {0.f, 0.f, 0.f, 0.f, 0.f, 0.f, 0.f, 0.f};
  const float* __restrict__ Arow = t3 + (size_t)(m0 + r) * HDIM;
  for (int k = 0; k < HDIM; k += 4) {
    const int ka = k + 2 * half;
    const float2 av = *(const float2*)(Arow + ka);
    v2f a; a[0] = av.x; a[1] = av.y;
    v2f b;
    b[0] = nvalid ? kw2[(size_t)ka * KDIM + r] : 0.f;
    b[1] = nvalid ? kw2[(size_t)(ka + 1) * KDIM + r] : 0.f;
    acc = __builtin_amdgcn_wmma_f32_16x16x4_f32(false, a, false, b,
                                                (short)0, acc, false, false);
  }
  if (nvalid) {
    const float bv = kb2[r];
#pragma unroll
    for (int v = 0; v < 8; ++v) {
      const int m = m0 + 8 * half + v;
      coeff_out[m * KDIM + r] = tanhf(acc[v] + bv);
    }
  }
}

// ---------------------------------------------------------------------------
// weight[b,o,i] = (base[o,i] + sum_k coeff[b,k]*bases[k,o,i]) * row[b,o] * col[b,i]
// Block tile = 8(o) x 128(i); each thread holds base + 8 bases float4s in
// registers (read ONCE globally, NT) and loops all 64 batches, streaming the
// 256 MB output with non-temporal b128 stores.  grid = 128*8 = 1024 blocks.
// ---------------------------------------------------------------------------
__global__ __launch_bounds__(256) void build_weight_kernel(
    const float* __restrict__ base, const float* __restrict__ bases,
    const float* __restrict__ row, const float* __restrict__ col,
    const float* __restrict__ coeff, float* __restrict__ weight) {
  const int bi = blockIdx.x;
  const int o0 = (bi >> 3) << 3;          // 128 o-tiles of 8
  const int i0 = (bi & 7) << 7;           // 8 i-tiles of 128
  const int t  = threadIdx.x;
  const int o  = o0 + (t >> 5);           // 32 threads per o-row
  const int i  = i0 + ((t & 31) << 2);    // float4 per thread

  const size_t oi = (size_t)o * IDIM + i;

  // Tile data: read exactly once from HBM (no reuse by other blocks) -> NT.
  const v4f b4 = __builtin_nontemporal_load((const v4f*)(base + oi));
  v4f bk[KDIM];
#pragma unroll
  for (int k = 0; k < KDIM; ++k)
    bk[k] = __builtin_nontemporal_load(
        (const v4f*)(bases + (size_t)k * ODIM * IDIM + oi));

  for (int b = 0; b < BDIM; ++b) {
    float ck[KDIM];
#pragma unroll
    for (int k = 0; k < KDIM; ++k) ck[k] = coeff[b * KDIM + k];

    v4f d = b4;
#pragma unroll
    for (int k = 0; k < KDIM; ++k) {
      d[0] = fmaf(ck[k], bk[k][0], d[0]);
      d[1] = fmaf(ck[k], bk[k][1], d[1]);
      d[2] = fmaf(ck[k], bk[k][2], d[2]);
      d[3] = fmaf(ck[k], bk[k][3], d[3]);
    }

    const float rv = row[(size_t)b * ODIM + o];
    const v4f cv = *(const v4f*)(col + (size_t)b * IDIM + i);
    d[0] *= rv * cv[0];
    d[1] *= rv * cv[1];
    d[2] *= rv * cv[2];
    d[3] *= rv * cv[3];

    __builtin_nontemporal_store(
        d, (v4f*)(weight + ((size_t)b << 20) + oi));  // 1<<20 = O*I
  }
}

// ---------------------------------------------------------------------------
// Host launch.  Inputs (setup_inputs order):
//  0:h 1:base 2:bases 3:rw1 4:rb1 5:rw2 6:rb2 7:cw1 8:cb1 9:cw2 10:cb2
//  11:kw1 12:kb1 13:kw2 14:kb2
// d_out = weight[64*1024*1024] | row[64*1024] | col[64*1024] | coeff[64*8]
// ---------------------------------------------------------------------------
extern "C" void kernel_launch(void* const* d_in, const int* in_sizes, int n_in,
                              void* d_out, int out_size, void* d_ws, size_t ws_size,
                              hipStream_t stream) {
  const float* h     = (const float*)d_in[0];
  const float* base  = (const float*)d_in[1];
  const float* bases = (const float*)d_in[2];
  const float* rw1 = (const float*)d_in[3];
  const float* rb1 = (const float*)d_in[4];
  const float* rw2 = (const float*)d_in[5];
  const float* rb2 = (const float*)d_in[6];
  const float* cw1 = (const float*)d_in[7];
  const float* cb1 = (const float*)d_in[8];
  const float* cw2 = (const float*)d_in[9];
  const float* cb2 = (const float*)d_in[10];
  const float* kw1 = (const float*)d_in[11];
  const float* kb1 = (const float*)d_in[12];
  const float* kw2 = (const float*)d_in[13];
  const float* kb2 = (const float*)d_in[14];

  float* out    = (float*)d_out;
  float* weight = out;                                    // [64,1024,1024]
  float* rowp   = out + (size_t)BDIM * ODIM * IDIM;       // [64,1024]
  float* colp   = rowp + (size_t)BDIM * ODIM;             // [64,1024]
  float* coeffp = colp + (size_t)BDIM * IDIM;             // [64,8]

  float* ws = (float*)d_ws;
  float* t1 = ws;                         // [64,1024]
  float* t2 = t1 + (size_t)BDIM * HDIM;   // [64,1024]
  float* t3 = t2 + (size_t)BDIM * HDIM;   // [64,1024]

  // Phase 1a: hidden activations of the three MLPs (WMMA f32).
  mlp_layer1_kernel<<<dim3(64, 1, 3), dim3(128), 0, stream>>>(
      h, rw1, rb1, cw1, cb1, kw1, kb1, t1, t2, t3);

  // Phase 1b: row / col heads (WMMA f32).
  mlp_layer2_kernel<<<dim3(64, 1, 2), dim3(128), 0, stream>>>(
      t1, t2, rw2, rb2, cw2, cb2, rowp, colp);

  // Phase 1c: coeff head, N=8 padded tile (WMMA f32).
  mlp_coeff_kernel<<<dim3(1), dim3(128), 0, stream>>>(t3, kw2, kb2, coeffp);

  // Phase 2: bandwidth-bound weight assembly (NT b128 streams).
  build_weight_kernel<<<dim3(1024), dim3(256), 0, stream>>>(
      base, bases, rowp, colp, coeffp, weight);
}